// PolyConvFrame_21414706938561
// MI455X (gfx1250) — compile-verified
//
#include <hip/hip_runtime.h>
#include <hip/hip_bf16.h>
#include <math.h>

// ---------------------------------------------------------------------------
// PolyConv / PowerConv: out[:,0,:] = x ; out[:,L,:] = tanh(alpha_L) * (A @ out[:,L-1,:])
// A = D^-1/2 W D^-1/2  (GCN normalization, deg over `row`, edge weights w)
//
// Memory-bound SpMM (~0.4 FLOP/B): one wave32 per edge; 32 lanes x float2 =
// 256B coalesced gather of h[col] (25.6MB feature slice is L2-resident in the
// 192MB L2), f32 atomics scatter to out[row]. Edge records (row,col,val) are
// async-staged into LDS with GLOBAL_LOAD_ASYNC_TO_LDS_B32, double buffered and
// fenced with s_wait_asynccnt — the gfx1250-specific data-movement path.
//
// The async builtins are used UNCONDITIONALLY in the device pass: a successful
// compile proves the async instructions lowered (no silent fallback).
// ---------------------------------------------------------------------------

#define D_FEAT 64
#define DEPTH 3
#define NODE_STRIDE (D_FEAT * (DEPTH + 1))   // 256 floats per node in out
#define SPMM_WAVES 8                          // waves per block (256 thr, wave32)
#define CHUNK 32                              // edges staged per wave per buffer

#if defined(__HIP_DEVICE_COMPILE__)
#define ASYNC_LDS 1                           // device pass: require the builtins
#else
#define ASYNC_LDS 0                           // host pass: builtins undeclared
#endif

__device__ __forceinline__ void async_copy_b32(const int* gsrc, int* ldst) {
#if ASYNC_LDS
  __builtin_amdgcn_global_load_async_to_lds_b32(
      (__attribute__((address_space(1))) int*)(unsigned long long)(uintptr_t)gsrc,
      (__attribute__((address_space(3))) int*)ldst,
      /*offset=*/0, /*cpol=*/0);
#else
  *ldst = *gsrc;
#endif
}

template <int N>
__device__ __forceinline__ void wait_async() {
#if ASYNC_LDS
  __builtin_amdgcn_s_wait_asynccnt(N);        // s_wait_asynccnt N
#endif
}

// ---------------------------------------------------------------------------
// Small prep kernels
// ---------------------------------------------------------------------------

__global__ __launch_bounds__(256) void zero_f32_kernel(float* __restrict__ p, int n) {
  int i = blockIdx.x * blockDim.x + threadIdx.x;
  if (i < n) p[i] = 0.0f;
}

__global__ __launch_bounds__(256) void deg_accum_kernel(const int* __restrict__ rows,
                                                        const float* __restrict__ w,
                                                        float* __restrict__ deg, int E) {
  int e = blockIdx.x * blockDim.x + threadIdx.x;
  if (e < E) atomicAdd(&deg[rows[e]], w[e]);
}

__global__ __launch_bounds__(256) void deg_to_dinv_kernel(float* __restrict__ deg, int n) {
  int i = blockIdx.x * blockDim.x + threadIdx.x;
  if (i < n) {
    float d = deg[i];
    deg[i] = (d > 0.0f) ? rsqrtf(d) : 0.0f;
  }
}

__global__ __launch_bounds__(256) void edge_norm_kernel(const int* __restrict__ rows,
                                                        const int* __restrict__ cols,
                                                        const float* __restrict__ w,
                                                        const float* __restrict__ dinv,
                                                        float* __restrict__ val, int E) {
  int e = blockIdx.x * blockDim.x + threadIdx.x;
  if (e < E) {
    // stream-ahead prefetch of the sequentially-consumed arrays (global_prefetch_b8)
    __builtin_prefetch(rows + e + 4096, 0, 0);
    __builtin_prefetch(cols + e + 4096, 0, 0);
    __builtin_prefetch(w + e + 4096, 0, 0);
    val[e] = dinv[rows[e]] * w[e] * dinv[cols[e]];
  }
}

// out[i,0,f] = x[i,f]; out[i,L,f] = 0 for L=1..3. One thread per (i,f).
__global__ __launch_bounds__(256) void init_out_kernel(const float* __restrict__ x,
                                                       float* __restrict__ out, int total) {
  int t = blockIdx.x * blockDim.x + threadIdx.x;
  if (t >= total) return;
  int i = t >> 6;          // node
  int f = t & 63;          // feature
  float xv = x[t];
  float* o = out + ((size_t)i * NODE_STRIDE) + f;
  o[0 * D_FEAT] = xv;
  o[1 * D_FEAT] = 0.0f;
  o[2 * D_FEAT] = 0.0f;
  o[3 * D_FEAT] = 0.0f;
}

// ---------------------------------------------------------------------------
// SpMM scatter layer: out[:,L,:] += a_L * val_e * out[col_e, L-1, :]
// One wave per 32-edge chunk, persistent chunk-stride loop, double-buffered
// async LDS staging of (row, col, val).
// ---------------------------------------------------------------------------

__device__ __forceinline__ void stage_chunk(int (&dst)[3][CHUNK],
                                            const int* __restrict__ rows,
                                            const int* __restrict__ cols,
                                            const int* __restrict__ valbits,
                                            int chunk, int nEdges, int lane) {
  int g = (chunk << 5) + lane;
  if (g >= nEdges) g = nEdges - 1;  // clamp; tail edges masked at consume time
  async_copy_b32(rows + g, &dst[0][lane]);
  async_copy_b32(cols + g, &dst[1][lane]);
  async_copy_b32(valbits + g, &dst[2][lane]);
}

__global__ __launch_bounds__(256) void spmm_layer_kernel(
    const int* __restrict__ rows, const int* __restrict__ cols,
    const float* __restrict__ val, const float* __restrict__ alphas,
    float* __restrict__ out, int layer, int nEdges) {
  __shared__ int lds[SPMM_WAVES][2][3][CHUNK];

  const int wave = threadIdx.x >> 5;
  const int lane = threadIdx.x & 31;
  const int wavesTotal = gridDim.x * SPMM_WAVES;
  const int nChunks = (nEdges + (CHUNK - 1)) >> 5;
  const float aL = tanhf(alphas[layer]);  // BASEALPHA = 1.0
  const int f0 = lane * 2;                // 32 lanes x 2 feats = 64 feats
  const size_t hOff = (size_t)(layer - 1) * D_FEAT + f0;
  const size_t oOff = (size_t)layer * D_FEAT + f0;
  const int* valbits = (const int*)val;

  int chunk0 = blockIdx.x * SPMM_WAVES + wave;
  if (chunk0 >= nChunks) return;

  stage_chunk(lds[wave][0], rows, cols, valbits, chunk0, nEdges, lane);
  int buf = 0;

  for (int chunk = chunk0; chunk < nChunks; chunk += wavesTotal) {
    const int next = chunk + wavesTotal;
    if (next < nChunks) {
      stage_chunk(lds[wave][buf ^ 1], rows, cols, valbits, next, nEdges, lane);
      wait_async<3>();  // <=3 outstanding -> current buffer's 3 loads landed
    } else {
      wait_async<0>();
    }

    const int base = chunk << 5;
    const int m = (nEdges - base < CHUNK) ? (nEdges - base) : CHUNK;
    const int(&cb)[3][CHUNK] = lds[wave][buf];

    for (int e = 0; e < m; ++e) {
      const int r = cb[0][e];                        // LDS broadcast reads
      const int c = cb[1][e];
      const float v = aL * __int_as_float(cb[2][e]); // fold a_L into edge val
      // 256B coalesced gather of h[col] (L2-resident working set)
      const float2 h = *(const float2*)(out + (size_t)c * NODE_STRIDE + hOff);
      float* dst = out + (size_t)r * NODE_STRIDE + oOff;
      atomicAdd(dst + 0, v * h.x);                   // global_atomic_add_f32
      atomicAdd(dst + 1, v * h.y);
    }
    buf ^= 1;
  }
}

// ---------------------------------------------------------------------------
// Launcher
// ---------------------------------------------------------------------------

extern "C" void kernel_launch(void* const* d_in, const int* in_sizes, int n_in,
                              void* d_out, int out_size, void* d_ws, size_t ws_size,
                              hipStream_t stream) {
  const float* x      = (const float*)d_in[0];   // [N, 64] f32
  const int*   eidx   = (const int*)d_in[1];     // [2, E] int
  const float* w      = (const float*)d_in[2];   // [E] f32
  const float* alphas = (const float*)d_in[3];   // [4] f32

  const int N = in_sizes[0] / D_FEAT;
  const int E = in_sizes[2];
  const int* rows = eidx;
  const int* cols = eidx + E;

  float* out = (float*)d_out;            // [N, 4, 64]
  float* deg = (float*)d_ws;             // N floats (becomes dinv in place)
  float* val = deg + N;                  // E floats

  const int TB = 256;
  const int nBlocksN = (N + TB - 1) / TB;
  const int nBlocksE = (E + TB - 1) / TB;

  // 1) deg = segment_sum(w, row)
  zero_f32_kernel<<<nBlocksN, TB, 0, stream>>>(deg, N);
  deg_accum_kernel<<<nBlocksE, TB, 0, stream>>>(rows, w, deg, E);
  // 2) dinv = deg>0 ? rsqrt(deg) : 0   (in place)
  deg_to_dinv_kernel<<<nBlocksN, TB, 0, stream>>>(deg, N);
  // 3) val_e = dinv[row] * w_e * dinv[col]
  edge_norm_kernel<<<nBlocksE, TB, 0, stream>>>(rows, cols, w, deg, val, E);
  // 4) out[:,0,:] = x ; out[:,1..3,:] = 0
  const int total = N * D_FEAT;
  init_out_kernel<<<(total + TB - 1) / TB, TB, 0, stream>>>(x, out, total);
  // 5) three SpMM layers (each needs the previous fully done -> stream order)
  const int spmmBlocks = 2048;  // 16384 persistent waves, chunk-stride over E/32 chunks
  for (int layer = 1; layer <= DEPTH; ++layer) {
    spmm_layer_kernel<<<spmmBlocks, TB, 0, stream>>>(rows, cols, val, alphas,
                                                     out, layer, E);
  }
}